// OutputNetwork_29317446763346
// MI455X (gfx1250) — compile-verified
//
#include <hip/hip_runtime.h>
#include <hip/hip_bf16.h>

typedef __attribute__((ext_vector_type(2))) float v2f;
typedef __attribute__((ext_vector_type(4))) float v4f;
typedef __attribute__((ext_vector_type(8))) float v8f;

#define HIDDEN     128
#define HALFH      64
#define STD_F      3.4f
#define MEAN_F     (-1.2f)
#define LDS_STRIDE 132   // 132 % 64 == 4 -> lanes 0..15 hit banks 4 apart: conflict-free

// SiLU with single-instruction v_rcp_f32 instead of the IEEE divide sequence.
__device__ __forceinline__ float fast_silu(float x) {
    float e = __expf(-x);                          // v_mul (log2e) + v_exp_f32
    return x * __builtin_amdgcn_rcpf(1.0f + e);    // v_add + v_rcp_f32 + v_mul
}

__global__ __launch_bounds__(256) void mlp_readout_wmma(
    const int*   __restrict__ z,
    const float* __restrict__ h,
    const int*   __restrict__ batch,
    const float* __restrict__ W1,
    const float* __restrict__ b1,
    const float* __restrict__ W2,
    const float* __restrict__ b2,
    float*       __restrict__ out)
{
    __shared__ float w1s[HALFH * LDS_STRIDE];

    const int tid = threadIdx.x;

    // Cooperative load of W1 (64x128 f32, row-major) into padded LDS.
    // 2048 float4 / 256 threads = 8 per thread, uniform (no divergence).
    const v4f* w1v = (const v4f*)W1;
    for (int i = tid; i < (HALFH * HIDDEN) / 4; i += 256) {
        int row  = i >> 5;            // 32 float4 per 128-wide row
        int col4 = (i & 31) << 2;
        *(v4f*)(&w1s[row * LDS_STRIDE + col4]) = w1v[i];
    }
    __syncthreads();

    const int lane = tid & 31;
    const int wave = tid >> 5;
    const int lm   = lane & 15;       // 0..15
    const int hi   = lane >> 4;       // 0 or 1

    const int tile     = blockIdx.x * 8 + wave;
    const int atomBase = tile * 16;

    // A fragment (16x4 f32): lane covers row m=lm; VGPR0 = K {kb | kb+2},
    // VGPR1 = K {kb+1 | kb+3} depending on lane half.
    const float* ap = h + (size_t)(atomBase + lm) * HIDDEN + 2 * hi;
    // B fragment (4x16 f32) from LDS: lane covers column n=lm (+16*t),
    // rows K follow the same half split.
    const float* bp = &w1s[(size_t)lm * LDS_STRIDE + 2 * hi];

    // Accumulators seeded with b1 broadcast: C layout has N = lm (+16*t) for
    // every VGPR of the tile, so a per-lane scalar splat is exact.
    v8f acc0, acc1, acc2, acc3;
    {
        float c0 = b1[lm], c1 = b1[lm + 16], c2 = b1[lm + 32], c3 = b1[lm + 48];
        #pragma unroll
        for (int r = 0; r < 8; ++r) { acc0[r] = c0; acc1[r] = c1; acc2[r] = c2; acc3[r] = c3; }
    }

    // Layer 1: y1[16x64] = h[16x128] @ W1^T[128x64] via 32 K-steps x 4 N-tiles
    #pragma unroll 4
    for (int kb = 0; kb < HIDDEN; kb += 4) {
        v2f a = *(const v2f*)(ap + kb);
        v2f bw;
        bw = *(const v2f*)(bp + kb);
        acc0 = __builtin_amdgcn_wmma_f32_16x16x4_f32(false, a, false, bw, (short)0, acc0, false, false);
        bw = *(const v2f*)(bp + 16 * LDS_STRIDE + kb);
        acc1 = __builtin_amdgcn_wmma_f32_16x16x4_f32(false, a, false, bw, (short)0, acc1, false, false);
        bw = *(const v2f*)(bp + 32 * LDS_STRIDE + kb);
        acc2 = __builtin_amdgcn_wmma_f32_16x16x4_f32(false, a, false, bw, (short)0, acc2, false, false);
        bw = *(const v2f*)(bp + 48 * LDS_STRIDE + kb);
        acc3 = __builtin_amdgcn_wmma_f32_16x16x4_f32(false, a, false, bw, (short)0, acc3, false, false);
    }

    // SiLU + layer 2 partial dot: per lane, N-columns {lm, lm+16, lm+32, lm+48}
    const float w20 = W2[lm], w21 = W2[lm + 16], w22 = W2[lm + 32], w23 = W2[lm + 48];
    float part[8];
    #pragma unroll
    for (int r = 0; r < 8; ++r) {
        float s0 = fast_silu(acc0[r]);
        float s1 = fast_silu(acc1[r]);
        float s2 = fast_silu(acc2[r]);
        float s3 = fast_silu(acc3[r]);
        part[r] = s0 * w20 + s1 * w21 + s2 * w22 + s3 * w23;
    }

    // Butterfly reduce over the 16 lanes of each half (N dimension).
    #pragma unroll
    for (int m = 8; m >= 1; m >>= 1) {
        #pragma unroll
        for (int r = 0; r < 8; ++r) part[r] += __shfl_xor(part[r], m, 16);
    }

    // Lower half VGPR r -> atom row r; upper half -> row r+8.
    // Lane (lm==r, lm<8) of each half finalizes one atom: 16 atoms -> 16 lanes.
    float yv = part[0];
    #pragma unroll
    for (int r = 1; r < 8; ++r) if (lm == r) yv = part[r];

    if (lm < 8) {
        int atom = atomBase + lm + hi * 8;
        float y  = (yv + b2[0]) * STD_F + MEAN_F;
        if (z[atom] > 0) atomicAdd(out + batch[atom], y);
    }
}

// Scalar tail for atom counts not divisible by 128 (empty for N=2,000,000).
__global__ void mlp_readout_tail(
    const int* __restrict__ z, const float* __restrict__ h,
    const int* __restrict__ batch,
    const float* __restrict__ W1, const float* __restrict__ b1,
    const float* __restrict__ W2, const float* __restrict__ b2,
    float* __restrict__ out, int start, int n)
{
    int i = blockIdx.x * blockDim.x + threadIdx.x;
    if (i >= n) return;
    int atom = start + i;
    float acc = b2[0];
    for (int j = 0; j < HALFH; ++j) {
        float d = b1[j];
        const float* hr = h + (size_t)atom * HIDDEN;
        const float* wr = W1 + (size_t)j * HIDDEN;
        for (int k = 0; k < HIDDEN; ++k) d += hr[k] * wr[k];
        float s = fast_silu(d);
        acc += s * W2[j];
    }
    float y = acc * STD_F + MEAN_F;
    if (z[atom] > 0) atomicAdd(out + batch[atom], y);
}

extern "C" void kernel_launch(void* const* d_in, const int* in_sizes, int n_in,
                              void* d_out, int out_size, void* d_ws, size_t ws_size,
                              hipStream_t stream) {
    const int*   z     = (const int*)  d_in[0];
    const float* hfeat = (const float*)d_in[1];
    // d_in[2] = pos (unused by the reference output)
    const int*   batch = (const int*)  d_in[3];
    const float* W1    = (const float*)d_in[4];
    const float* b1    = (const float*)d_in[5];
    const float* W2    = (const float*)d_in[6];
    const float* b2    = (const float*)d_in[7];
    float* out = (float*)d_out;

    const int nAtoms = in_sizes[0];

    // out is poisoned by the harness and accumulated via atomics: zero it each call.
    hipMemsetAsync(out, 0, (size_t)out_size * sizeof(float), stream);

    const int atomsPerBlock = 128;                 // 8 waves x 16 atoms
    const int blocks = nAtoms / atomsPerBlock;
    if (blocks > 0) {
        mlp_readout_wmma<<<blocks, 256, 0, stream>>>(z, hfeat, batch, W1, b1, W2, b2, out);
    }
    const int done = blocks * atomsPerBlock;
    const int rem  = nAtoms - done;
    if (rem > 0) {
        mlp_readout_tail<<<(rem + 127) / 128, 128, 0, stream>>>(
            z, hfeat, batch, W1, b1, W2, b2, out, done, rem);
    }
}